// graph_transf_block_17497696764590
// MI455X (gfx1250) — compile-verified
//
#include <hip/hip_runtime.h>
#include <math.h>

#define NODES 50000
#define NEDGE 800000
#define DIM   128

typedef float v2f __attribute__((ext_vector_type(2)));
typedef float v8f __attribute__((ext_vector_type(8)));

// ---------------------------------------------------------------------------
// Fused 4-output GEMM:  O_w = X @ W_w + b_w  for w in {q,k,v,skip}
// Block = 128 threads = 4 waves. Wave w computes the 16x16 tile
// (blockIdx.x*16 rows, blockIdx.y*16 cols) of output matrix w via
// V_WMMA_F32_16X16X4_F32 with K accumulated over 32 steps of 4.
//
// fp32 16x4 A layout (ISA 7.12.2): lanes 0-15 hold M=lane, VGPR0=K0,VGPR1=K1;
// lanes 16-31 hold M=lane-16, VGPR0=K2,VGPR1=K3  -> one float2 load per lane.
// B (4x16) mirrored: lane%16 = N, lanes>=16 carry K=2,3 rows.
// C/D 16x16 f32: VGPR r -> M = r + 8*(lane>=16), N = lane%16.
// ---------------------------------------------------------------------------
__global__ __launch_bounds__(128)
void gemm4_kernel(const float* __restrict__ X,
                  const float* __restrict__ W0, const float* __restrict__ B0,
                  const float* __restrict__ W1, const float* __restrict__ B1,
                  const float* __restrict__ W2, const float* __restrict__ B2,
                  const float* __restrict__ W3, const float* __restrict__ B3,
                  float* __restrict__ O0, float* __restrict__ O1,
                  float* __restrict__ O2, float* __restrict__ O3)
{
    const int wid  = threadIdx.x >> 5;        // wave id 0..3 -> which matrix
    const int lane = threadIdx.x & 31;
    const int lm   = lane & 15;               // M (for A) / N (for B,C,D)
    const int kh   = (lane >> 4) << 1;        // K sub-offset: 0 or 2
    const int mrow = blockIdx.x << 4;
    const int ncol = blockIdx.y << 4;

    const float* W = (wid == 0) ? W0 : (wid == 1) ? W1 : (wid == 2) ? W2 : W3;
    const float* B = (wid == 0) ? B0 : (wid == 1) ? B1 : (wid == 2) ? B2 : B3;
    float*       O = (wid == 0) ? O0 : (wid == 1) ? O1 : (wid == 2) ? O2 : O3;

    const float* arow = X + (size_t)(mrow + lm) * DIM;
    const float* bcol = W + ncol + lm;

    v8f acc = {};
#pragma unroll
    for (int k0 = 0; k0 < DIM; k0 += 4) {
        v2f a = *(const v2f*)(arow + k0 + kh);          // A[m][k0+kh], A[m][k0+kh+1]
        v2f b;
        b.x = bcol[(size_t)(k0 + kh)     * DIM];        // W[k0+kh  ][n]
        b.y = bcol[(size_t)(k0 + kh + 1) * DIM];        // W[k0+kh+1][n]
        acc = __builtin_amdgcn_wmma_f32_16x16x4_f32(false, a, false, b,
                                                    (short)0, acc, false, false);
    }

    const float bias  = B[ncol + lm];
    const int   mbase = mrow + ((lane >> 4) << 3);
#pragma unroll
    for (int r = 0; r < 8; ++r)
        O[(size_t)(mbase + r) * DIM + ncol + lm] = acc[r] + bias;
}

// ---------------------------------------------------------------------------
// Segment-softmax state init: amax (ordered-uint encoding, 0 = -inf), denom=0
// ---------------------------------------------------------------------------
__global__ __launch_bounds__(256)
void init_seg_kernel(unsigned* __restrict__ amaxb, float* __restrict__ denom)
{
    int i = blockIdx.x * blockDim.x + threadIdx.x;
    if (i < NODES) { amaxb[i] = 0u; denom[i] = 0.0f; }
}

// ordered encoding for float atomicMax via uint
__device__ __forceinline__ unsigned enc_ord(float f) {
    unsigned u = __float_as_uint(f);
    return (u & 0x80000000u) ? ~u : (u | 0x80000000u);
}
__device__ __forceinline__ float dec_ord(unsigned u) {
    unsigned b = (u & 0x80000000u) ? (u & 0x7fffffffu) : ~u;
    return __uint_as_float(b);
}

// ---------------------------------------------------------------------------
// Per-edge logits + segment max. One wave per edge; float4/lane dot, shfl tree.
// ---------------------------------------------------------------------------
__global__ __launch_bounds__(256)
void edge_logits_kernel(const float* __restrict__ q, const float* __restrict__ k,
                        const int* __restrict__ src, const int* __restrict__ dst,
                        float* __restrict__ alpha, unsigned* __restrict__ amaxb)
{
    const int wave = (blockIdx.x * blockDim.x + threadIdx.x) >> 5;
    const int lane = threadIdx.x & 31;
    if (wave >= NEDGE) return;
    const int s = src[wave], d = dst[wave];

    const float4 qv = *(const float4*)(q + (size_t)d * DIM + lane * 4);
    const float4 kv = *(const float4*)(k + (size_t)s * DIM + lane * 4);
    float p = qv.x * kv.x + qv.y * kv.y + qv.z * kv.z + qv.w * kv.w;
#pragma unroll
    for (int off = 16; off > 0; off >>= 1) p += __shfl_xor(p, off, 32);

    if (lane == 0) {
        const float a = p * 0.08838834764831845f;   // 1/sqrt(128)
        alpha[wave] = a;
        atomicMax(amaxb + d, enc_ord(a));
    }
}

// ---------------------------------------------------------------------------
// exp(alpha - amax[dst]) in place + segment-sum of denom.
// (Every dst read here has >=1 incoming edge, so amax is always finite.)
// ---------------------------------------------------------------------------
__global__ __launch_bounds__(256)
void edge_softmax_kernel(float* alpha, const int* __restrict__ dst,
                         const unsigned* __restrict__ amaxb,
                         float* __restrict__ denom)
{
    int e = blockIdx.x * blockDim.x + threadIdx.x;
    if (e >= NEDGE) return;
    const int d = dst[e];
    const float m  = dec_ord(amaxb[d]);
    const float ex = expf(alpha[e] - m);
    alpha[e] = ex;
    atomicAdd(denom + d, ex);
}

// ---------------------------------------------------------------------------
// out[dst] += v[src] * (ex / (denom[dst] + 1e-16)). One wave per edge.
// ---------------------------------------------------------------------------
__global__ __launch_bounds__(256)
void edge_aggregate_kernel(const float* __restrict__ ex, const float* __restrict__ v,
                           const int* __restrict__ src, const int* __restrict__ dst,
                           const float* __restrict__ denom, float* __restrict__ out)
{
    const int wave = (blockIdx.x * blockDim.x + threadIdx.x) >> 5;
    const int lane = threadIdx.x & 31;
    if (wave >= NEDGE) return;
    const int s = src[wave], d = dst[wave];
    const float w = ex[wave] / (denom[d] + 1e-16f);

    const float4 vv = *(const float4*)(v + (size_t)s * DIM + lane * 4);
    float* o = out + (size_t)d * DIM + lane * 4;
    atomicAdd(o + 0, vv.x * w);
    atomicAdd(o + 1, vv.y * w);
    atomicAdd(o + 2, vv.z * w);
    atomicAdd(o + 3, vv.w * w);
}

// ---------------------------------------------------------------------------
__global__ __launch_bounds__(256)
void elu_inplace_kernel(float* h, int n)
{
    int i = blockIdx.x * blockDim.x + threadIdx.x;
    if (i < n) { float x = h[i]; h[i] = (x > 0.0f) ? x : expm1f(x); }
}

// ---------------------------------------------------------------------------
static void run_layer(const float* X,
                      const float* Wq, const float* bq,
                      const float* Wk, const float* bk,
                      const float* Wv, const float* bv,
                      const float* Ws, const float* bs,
                      float* q, float* k, float* v, float* out,
                      float* alpha, unsigned* amaxb, float* denom,
                      const int* esrc, const int* edst, hipStream_t stream)
{
    dim3 g(NODES / 16, DIM / 16);                      // 3125 x 8, exact
    gemm4_kernel<<<g, 128, 0, stream>>>(X, Wq, bq, Wk, bk, Wv, bv, Ws, bs,
                                        q, k, v, out);
    init_seg_kernel<<<(NODES + 255) / 256, 256, 0, stream>>>(amaxb, denom);
    edge_logits_kernel<<<(NEDGE * 32 + 255) / 256, 256, 0, stream>>>(
        q, k, esrc, edst, alpha, amaxb);
    edge_softmax_kernel<<<(NEDGE + 255) / 256, 256, 0, stream>>>(
        alpha, edst, amaxb, denom);
    edge_aggregate_kernel<<<(NEDGE * 32 + 255) / 256, 256, 0, stream>>>(
        alpha, v, esrc, edst, denom, out);
}

extern "C" void kernel_launch(void* const* d_in, const int* in_sizes, int n_in,
                              void* d_out, int out_size, void* d_ws, size_t ws_size,
                              hipStream_t stream)
{
    (void)in_sizes; (void)n_in; (void)out_size; (void)ws_size;

    const float* x    = (const float*)d_in[0];
    const int*   esrc = (const int*)d_in[1];
    const int*   edst = (const int*)d_in[2];
    const float* W1q = (const float*)d_in[3],  *b1q = (const float*)d_in[4];
    const float* W1k = (const float*)d_in[5],  *b1k = (const float*)d_in[6];
    const float* W1v = (const float*)d_in[7],  *b1v = (const float*)d_in[8];
    const float* W1s = (const float*)d_in[9],  *b1s = (const float*)d_in[10];
    const float* W2q = (const float*)d_in[11], *b2q = (const float*)d_in[12];
    const float* W2k = (const float*)d_in[13], *b2k = (const float*)d_in[14];
    const float* W2v = (const float*)d_in[15], *b2v = (const float*)d_in[16];
    const float* W2s = (const float*)d_in[17], *b2s = (const float*)d_in[18];

    const size_t ND = (size_t)NODES * DIM;             // 6.4M floats
    float* ws    = (float*)d_ws;
    float* q     = ws;
    float* k     = ws + ND;
    float* v     = ws + 2 * ND;
    float* h     = ws + 3 * ND;                        // layer-1 output / layer-2 input
    float* alpha = ws + 4 * ND;                        // NEDGE floats
    unsigned* amaxb = (unsigned*)(alpha + NEDGE);      // NODES
    float* denom = (float*)(amaxb + NODES);            // NODES

    // Layer 1: h = TransformerConv(x)
    run_layer(x, W1q, b1q, W1k, b1k, W1v, b1v, W1s, b1s,
              q, k, v, h, alpha, amaxb, denom, esrc, edst, stream);
    elu_inplace_kernel<<<((int)ND + 255) / 256, 256, 0, stream>>>(h, (int)ND);

    // Layer 2: out = TransformerConv(elu(h))
    run_layer(h, W2q, b2q, W2k, b2k, W2v, b2v, W2s, b2s,
              q, k, v, (float*)d_out, alpha, amaxb, denom, esrc, edst, stream);
}